// SoftHistLoss_53704271069241
// MI455X (gfx1250) — compile-verified
//
#include <hip/hip_runtime.h>

// ---------------------------------------------------------------------------
// Soft-histogram L1 loss for MI455X (gfx1250, wave32).
//
// Roofline: 100 MB of input @ 23.3 TB/s => ~4.3 us floor. The sigmoid math is
// the only real compute: via the telescoping edge-sum formulation we need
// 11 exp + 11 rcp (TRANS) per pixel instead of the naive 20+20, and TRANS
// co-executes with the VALU fma/add stream on CDNA5. Cross-thread reduction
// is done with V_WMMA_F32_16X16X4_F32 (ones-matrix column-sum trick) in full
// fp32 precision.
// ---------------------------------------------------------------------------

#define NB            11          // histogram edges 0..10
#define NBINS         10
#define NSLICES       48          // 16 batches * 3 channels
#define SLICE_ELEMS   (512 * 512) // 262144 pixels per (b,c) slice
#define BLKS_PER_SL   16
#define THREADS       256
#define VEC           4
#define ITERS         (SLICE_ELEMS / (BLKS_PER_SL * THREADS * VEC)) // = 16

// m = -log2(e) * 150 * v + log2(e) * 15 * k   ==>  sigma = 1/(1+exp2(m))
#define MCOEF  (-216.40425613334451f)   // -150*log2(e)
#define KCOEF  (21.640425613334451f)    //   15*log2(e)

typedef __attribute__((ext_vector_type(2))) float v2f;
typedef __attribute__((ext_vector_type(8))) float v8f;

__global__ void zero_ws_kernel(float* __restrict__ ws) {
    for (int i = threadIdx.x; i < 2 * NSLICES * 16; i += blockDim.x)
        ws[i] = 0.0f;
}

__global__ __launch_bounds__(THREADS)
void soft_hist_kernel(const float* __restrict__ x,
                      const float* __restrict__ y,
                      float* __restrict__ ws) {
    const int bid   = blockIdx.x;
    const int img   = bid / (NSLICES * BLKS_PER_SL);
    const int rem   = bid % (NSLICES * BLKS_PER_SL);
    const int slice = rem / BLKS_PER_SL;
    const int chunk = rem % BLKS_PER_SL;
    const int tid   = threadIdx.x;

    const float* __restrict__ src =
        (img == 0 ? x : y) + (size_t)slice * SLICE_ELEMS
                           + (size_t)chunk * (SLICE_ELEMS / BLKS_PER_SL);

    float E[NB];
#pragma unroll
    for (int k = 0; k < NB; ++k) E[k] = 0.0f;

    // ---- hot loop: streaming float4 loads, 11 edge sigmoids per pixel ----
    for (int it = 0; it < ITERS; ++it) {
        const float4 vv = *(const float4*)(src + (size_t)it * THREADS * VEC + tid * VEC);
        if (it + 1 < ITERS)
            __builtin_prefetch(src + (size_t)(it + 1) * THREADS * VEC + tid * VEC, 0, 0);

        float vals[VEC] = {vv.x, vv.y, vv.z, vv.w};
#pragma unroll
        for (int p = 0; p < VEC; ++p) {
            const float v = vals[p];
#pragma unroll
            for (int k = 0; k < NB; ++k) {
                float m = fmaf(v, MCOEF, KCOEF * (float)k); // -t*log2(e)
                float u = __builtin_amdgcn_exp2f(m);        // v_exp_f32 (TRANS)
                E[k] += __builtin_amdgcn_rcpf(1.0f + u);    // v_rcp_f32 (TRANS)
            }
        }
    }

    // ---- block reduction: LDS -> WMMA column sums (fp32, 16x16x4) --------
    __shared__ float sacc[THREADS * 16];
#pragma unroll
    for (int k = 0; k < NB; ++k)    sacc[tid * 16 + k] = E[k];
#pragma unroll
    for (int k = NB; k < 16; ++k)   sacc[tid * 16 + k] = 0.0f;
    __syncthreads();

    if (tid < 32) {                      // one full wave, EXEC == all ones
        const int lane = tid;
        const int col  = lane & 15;
        const int rs   = (lane < 16) ? 0 : 2;

        v2f a; a.x = 1.0f; a.y = 1.0f;   // ones A-matrix => D = column sums of B
        v8f c = {};
#pragma unroll 8
        for (int i = 0; i < THREADS / 4; ++i) {
            v2f b;
            b.x = sacc[(4 * i + rs + 0) * 16 + col];
            b.y = sacc[(4 * i + rs + 1) * 16 + col];
            // D = A(16x4 ones) * B(4x16 partials) + C  -> v_wmma_f32_16x16x4_f32
            c = __builtin_amdgcn_wmma_f32_16x16x4_f32(
                    false, a, false, b, (short)0, c, false, false);
        }
        // D VGPR0: lanes 0..15 hold (M=0, N=lane) => lane < NB owns edge sum
        if (lane < NB)
            atomicAdd(&ws[(img * NSLICES + slice) * 16 + lane], c[0]);
    }
}

__global__ void finalize_kernel(const float* __restrict__ ws,
                                float* __restrict__ out) {
    __shared__ float red[64];
    const int t = threadIdx.x;
    float acc = 0.0f;
    if (t < NSLICES) {
        const float* Ex = ws + t * 16;
        const float* Ey = ws + (NSLICES + t) * 16;
        float s = 0.0f;
#pragma unroll
        for (int j = 0; j < NBINS; ++j) {
            float hx = Ex[j] - Ex[j + 1];   // telescoped bin weights
            float hy = Ey[j] - Ey[j + 1];
            s += fabsf(hx - hy);
        }
        acc = s * (1.0f / NBINS);           // mean over bins
    }
    red[t] = acc;
    __syncthreads();
    for (int off = 32; off > 0; off >>= 1) {
        if (t < off) red[t] += red[t + off];
        __syncthreads();
    }
    if (t == 0) out[0] = red[0] * (0.0001f / 16.0f);  // sum/B * 1e-4
}

extern "C" void kernel_launch(void* const* d_in, const int* in_sizes, int n_in,
                              void* d_out, int out_size, void* d_ws, size_t ws_size,
                              hipStream_t stream) {
    const float* x = (const float*)d_in[0];
    const float* y = (const float*)d_in[1];
    float* ws  = (float*)d_ws;
    float* out = (float*)d_out;

    zero_ws_kernel<<<1, 256, 0, stream>>>(ws);
    soft_hist_kernel<<<2 * NSLICES * BLKS_PER_SL, THREADS, 0, stream>>>(x, y, ws);
    finalize_kernel<<<1, 64, 0, stream>>>(ws, out);
}